// TrajectoryAttention_41566693491459
// MI455X (gfx1250) — compile-verified
//
#include <hip/hip_runtime.h>
#include <hip/hip_bf16.h>

// ---------------------------------------------------------------------------
// TrajectoryAttention for MI455X (gfx1250, wave32, WMMA fp32 16x16x4).
// B=2, F=8, P=196, C=768, H=12, d=64, N=1569, S=1568.
// Fused spatial attention avoids materializing the 236MB qk tensor.
// GEMM uses 32x64 tiles per wave: 8 WMMAs per (2 A + 8 B) loads.
// ---------------------------------------------------------------------------

constexpr int   B_ = 2;
constexpr int   F_ = 8;
constexpr int   P_ = 196;
constexpr int   C_ = 768;
constexpr int   H_ = 12;
constexpr int   N_ = 1569;   // 1 + F*P
constexpr int   S_ = 1568;   // F*P
constexpr float SCALE_ = 0.125f;  // 64^-0.5

typedef float v2f __attribute__((ext_vector_type(2)));
typedef float v8f __attribute__((ext_vector_type(8)));

__device__ __forceinline__ v8f wmma4(v2f a, v2f b, v8f c) {
  // D = A(16x4,f32) * B(4x16,f32) + C(16x16,f32)
  return __builtin_amdgcn_wmma_f32_16x16x4_f32(
      /*neg_a=*/false, a, /*neg_b=*/false, b,
      /*c_mod=*/(short)0, c, /*reuse_a=*/false, /*reuse_b=*/false);
}

// ---------------------------------------------------------------------------
// Generic fp32 WMMA GEMM: C[M,Nc] = A[M,K](lda) * B[K,Nc](ldb) (+bias).
// One wave computes a 32x64 tile (2 A-fragments x 4 B-subtiles = 8 accums;
// each B fragment is reused across both row-halves). Nc must be a multiple
// of 64, K a multiple of 4; ragged M handled by clamping loads and guarding
// stores.
// ---------------------------------------------------------------------------
__global__ __launch_bounds__(256) void gemm_wmma_f32(
    const float* __restrict__ A, int lda,
    const float* __restrict__ Bm, int ldb,
    float* __restrict__ Cm, int ldc,
    int M, int Nc, int K,
    const float* __restrict__ bias)
{
  const int lane = threadIdx.x & 31;
  const int wave = threadIdx.x >> 5;
  const int tilesN = Nc >> 6;
  const int tilesM = (M + 31) >> 5;
  const int tile = blockIdx.x * 8 + wave;
  if (tile >= tilesM * tilesN) return;          // wave-uniform exit
  const int tm = tile / tilesN;
  const int tn = tile - tm * tilesN;
  const int row0 = tm << 5;
  const int col0 = tn << 6;

  const int m  = lane & 15;
  const int kh = (lane >> 4) << 1;              // 0 or 2: K-pair select
  int ar0 = row0 + m;      if (ar0 >= M) ar0 = M - 1;
  int ar1 = row0 + 16 + m; if (ar1 >= M) ar1 = M - 1;
  const float* ap0 = A + (size_t)ar0 * lda + kh;
  const float* ap1 = A + (size_t)ar1 * lda + kh;
  const float* bp0 = Bm + (size_t)kh * ldb + col0 + m;

  v8f c00{}, c01{}, c02{}, c03{};
  v8f c10{}, c11{}, c12{}, c13{};
  #pragma unroll 2
  for (int k0 = 0; k0 < K; k0 += 4) {
    v2f a0 = *(const v2f*)(ap0 + k0);
    v2f a1 = *(const v2f*)(ap1 + k0);
    const float* bp = bp0 + (size_t)k0 * ldb;
    v2f b0 = { bp[0],  bp[ldb] };
    v2f b1 = { bp[16], bp[16 + ldb] };
    v2f b2 = { bp[32], bp[32 + ldb] };
    v2f b3 = { bp[48], bp[48 + ldb] };
    c00 = wmma4(a0, b0, c00);
    c10 = wmma4(a1, b0, c10);
    c01 = wmma4(a0, b1, c01);
    c11 = wmma4(a1, b1, c11);
    c02 = wmma4(a0, b2, c02);
    c12 = wmma4(a1, b2, c12);
    c03 = wmma4(a0, b3, c03);
    c13 = wmma4(a1, b3, c13);
  }

  const int rr = (lane >> 4) << 3;              // row offset 0 or 8
  const int cc = lane & 15;
  float b0v = 0.f, b1v = 0.f, b2v = 0.f, b3v = 0.f;
  if (bias) {
    b0v = bias[col0 + cc];      b1v = bias[col0 + 16 + cc];
    b2v = bias[col0 + 32 + cc]; b3v = bias[col0 + 48 + cc];
  }
  #pragma unroll
  for (int r = 0; r < 8; ++r) {
    const int rowA = row0 + rr + r;
    if (rowA < M) {
      float* op = Cm + (size_t)rowA * ldc + col0 + cc;
      op[0]  = c00[r] + b0v;
      op[16] = c01[r] + b1v;
      op[32] = c02[r] + b2v;
      op[48] = c03[r] + b3v;
    }
    const int rowB = row0 + 16 + rr + r;
    if (rowB < M) {
      float* op = Cm + (size_t)rowB * ldc + col0 + cc;
      op[0]  = c10[r] + b0v;
      op[16] = c11[r] + b1v;
      op[32] = c12[r] + b2v;
      op[48] = c13[r] + b3v;
    }
  }
}

// ---------------------------------------------------------------------------
// CLS-token attention: per (b,h): softmax(scale*q0.K^T) @ V over all N tokens.
// Writes row 0 of out_pre for each batch.
// ---------------------------------------------------------------------------
__global__ __launch_bounds__(256) void cls_attn_kernel(
    const float* __restrict__ qkv, float* __restrict__ outp)
{
  const int bh = blockIdx.x;
  const int b = bh / H_, h = bh % H_;
  const int tid = threadIdx.x;
  __shared__ float qsh[64];
  __shared__ float logits[N_];
  __shared__ float red[256];

  const float* q0 = qkv + (size_t)(b * N_) * (3 * C_) + h * 64;
  if (tid < 64) qsh[tid] = q0[tid] * SCALE_;
  __syncthreads();

  const float* kbase = qkv + (size_t)(b * N_) * (3 * C_) + C_ + h * 64;
  float lmax = -1e30f;
  for (int j = tid; j < N_; j += 256) {
    const float* kr = kbase + (size_t)j * (3 * C_);
    float dot = 0.f;
    #pragma unroll 8
    for (int dd = 0; dd < 64; ++dd) dot += qsh[dd] * kr[dd];
    logits[j] = dot;
    lmax = fmaxf(lmax, dot);
  }
  red[tid] = lmax; __syncthreads();
  for (int s = 128; s; s >>= 1) {
    if (tid < s) red[tid] = fmaxf(red[tid], red[tid + s]);
    __syncthreads();
  }
  const float mx = red[0];
  __syncthreads();
  float lsum = 0.f;
  for (int j = tid; j < N_; j += 256) {
    float e = __expf(logits[j] - mx);
    logits[j] = e; lsum += e;
  }
  red[tid] = lsum; __syncthreads();
  for (int s = 128; s; s >>= 1) {
    if (tid < s) red[tid] += red[tid + s];
    __syncthreads();
  }
  const float inv = 1.0f / red[0];
  __syncthreads();
  if (tid < 64) {
    const float* vb = qkv + (size_t)(b * N_) * (3 * C_) + 2 * C_ + h * 64 + tid;
    float acc = 0.f;
    for (int j = 0; j < N_; ++j) acc += logits[j] * vb[(size_t)j * (3 * C_)];
    outp[(size_t)(b * N_) * C_ + h * 64 + tid] = acc * inv;
  }
}

// ---------------------------------------------------------------------------
// Fused spatial attention. One wave per (q-tile of 16, frame f, b*h):
//   scores(16x196) = scale * Q(16x64) @ Kf(196x64)^T     [WMMA, 13 N-tiles]
//   softmax over 196 (in LDS, wave shfl reductions)
//   out(16x64)    = P(16x196) @ Vf(196x64)               [WMMA, K=49*4]
// Result goes to x4[b, s, f, h*64 + :].
// ---------------------------------------------------------------------------
#define LSROW 212   // padded LDS row stride (floats): even + conflict-free

__global__ __launch_bounds__(32) void spatial_attn_kernel(
    const float* __restrict__ qkv, float* __restrict__ x4)
{
  const int qt = blockIdx.x;            // 0..97  (S/16)
  const int f  = blockIdx.y;            // 0..7
  const int bh = blockIdx.z;            // 0..23
  const int b = bh / H_, h = bh % H_;
  const int lane = threadIdx.x;
  const int m  = lane & 15;
  const int kh = (lane >> 4) << 1;

  __shared__ float sc[16 * LSROW];
  __shared__ float rinv[16];

  // Preload 16 A-fragments of Q (16 rows x 64 dims).
  const int sQ = qt * 16 + m;
  const float* qrow = qkv + (size_t)(b * N_ + 1 + sQ) * (3 * C_) + h * 64;
  v2f aq[16];
  #pragma unroll
  for (int t = 0; t < 16; ++t)
    aq[t] = *(const v2f*)(qrow + 4 * t + kh);

  // Phase 1: scores into LDS.
  const float* kbase = qkv + (size_t)(b * N_ + 1 + f * P_) * (3 * C_) + C_ + h * 64;
  const int rr = (lane >> 4) << 3;
  for (int jt = 0; jt < 13; ++jt) {
    int j = jt * 16 + m; if (j > P_ - 1) j = P_ - 1;   // clamp ragged tail
    const float* kp = kbase + (size_t)j * (3 * C_) + kh;
    v8f acc{};
    #pragma unroll
    for (int t = 0; t < 16; ++t) {
      v2f bfrag = *(const v2f*)(kp + 4 * t);
      acc = wmma4(aq[t], bfrag, acc);
    }
    #pragma unroll
    for (int r = 0; r < 8; ++r)
      sc[(rr + r) * LSROW + jt * 16 + m] = acc[r] * SCALE_;
  }

  // Phase 2: softmax over the 196 valid columns of each of 16 rows.
  for (int r = 0; r < 16; ++r) {
    float mx = -1e30f;
    for (int j = lane; j < P_; j += 32) mx = fmaxf(mx, sc[r * LSROW + j]);
    #pragma unroll
    for (int off = 16; off; off >>= 1) mx = fmaxf(mx, __shfl_xor(mx, off, 32));
    float sum = 0.f;
    for (int j = lane; j < P_; j += 32) {
      float e = __expf(sc[r * LSROW + j] - mx);
      sc[r * LSROW + j] = e;
      sum += e;
    }
    #pragma unroll
    for (int off = 16; off; off >>= 1) sum += __shfl_xor(sum, off, 32);
    if (lane == 0) rinv[r] = 1.0f / sum;
  }

  // Phase 3: out = P @ Vf  (K = 196 = 49*4 exactly), normalize on store.
  const float* vbase = qkv + (size_t)(b * N_ + 1 + f * P_) * (3 * C_) + 2 * C_ + h * 64;
  v8f o0{}, o1{}, o2{}, o3{};
  for (int t = 0; t < 49; ++t) {
    const int k0 = 4 * t;
    v2f a = *(const v2f*)(&sc[m * LSROW + k0 + kh]);
    const float* vp = vbase + (size_t)(k0 + kh) * (3 * C_) + m;
    v2f b0 = { vp[0],  vp[3 * C_] };
    v2f b1 = { vp[16], vp[16 + 3 * C_] };
    v2f b2 = { vp[32], vp[32 + 3 * C_] };
    v2f b3 = { vp[48], vp[48 + 3 * C_] };
    o0 = wmma4(a, b0, o0);
    o1 = wmma4(a, b1, o1);
    o2 = wmma4(a, b2, o2);
    o3 = wmma4(a, b3, o3);
  }
  #pragma unroll
  for (int r = 0; r < 8; ++r) {
    const int row = qt * 16 + rr + r;
    const float inv = rinv[rr + r];
    float* op = x4 + (((size_t)(b * S_ + row) * F_ + f) * C_) + h * 64 + m;
    op[0]  = o0[r] * inv;
    op[16] = o1[r] * inv;
    op[32] = o2[r] * inv;
    op[48] = o3[r] * inv;
  }
}

// ---------------------------------------------------------------------------
// xd[b,s,c] = x4[b, s, s/P, c]  (the frame-diagonal).
// ---------------------------------------------------------------------------
__global__ void gather_xd_kernel(const float* __restrict__ x4,
                                 float* __restrict__ xd)
{
  const size_t i = (size_t)blockIdx.x * blockDim.x + threadIdx.x;
  const size_t total = (size_t)B_ * S_ * C_;
  if (i >= total) return;
  const int c = (int)(i % C_);
  const size_t bs = i / C_;
  const int s = (int)(bs % S_);
  const int fr = s / P_;
  xd[i] = x4[((bs * F_) + fr) * C_ + c];
}

// ---------------------------------------------------------------------------
// Second-stage attention over frames: per (b,s):
//   logits[h,f] = scale * <q2[b,s,h], k2[b,s,f,h]>,   softmax over f (F=8)
//   attn2 written to its d_out slice; out_pre row 1+s = sum_f p * x4[b,s,f,:].
// ---------------------------------------------------------------------------
__global__ __launch_bounds__(96) void second_attn_kernel(
    const float* __restrict__ q2, const float* __restrict__ k2,
    const float* __restrict__ x4, float* __restrict__ outp,
    float* __restrict__ attn2)
{
  const int bs = blockIdx.x;            // 0..B*S-1
  const int b = bs / S_, s = bs % S_;
  const int tid = threadIdx.x;          // 96 = 12 heads x 8 frames
  const int h = tid >> 3, f = tid & 7;
  __shared__ float pr[H_][F_];

  {
    const float* qr = q2 + (size_t)bs * C_ + h * 64;
    const float* kr = k2 + ((size_t)bs * F_ + f) * C_ + h * 64;
    float dot = 0.f;
    #pragma unroll 8
    for (int dd = 0; dd < 64; ++dd) dot += qr[dd] * kr[dd];
    pr[h][f] = dot * SCALE_;
  }
  __syncthreads();

  float mx = -1e30f;
  #pragma unroll
  for (int ff = 0; ff < F_; ++ff) mx = fmaxf(mx, pr[h][ff]);
  float sum = 0.f;
  #pragma unroll
  for (int ff = 0; ff < F_; ++ff) sum += __expf(pr[h][ff] - mx);
  const float p = __expf(pr[h][f] - mx) / sum;
  __syncthreads();
  pr[h][f] = p;
  attn2[(((size_t)b * H_ + h) * S_ + s) * F_ + f] = p;
  __syncthreads();

  for (int c = tid; c < C_; c += 96) {
    const int hc = c >> 6;
    const float* xr = x4 + ((size_t)bs * F_) * C_ + c;
    float acc = 0.f;
    #pragma unroll
    for (int ff = 0; ff < F_; ++ff) acc += pr[hc][ff] * xr[(size_t)ff * C_];
    outp[(size_t)(b * N_ + 1 + s) * C_ + c] = acc;
  }
}

// ---------------------------------------------------------------------------
// Host-side orchestration.
// ---------------------------------------------------------------------------
extern "C" void kernel_launch(void* const* d_in, const int* in_sizes, int n_in,
                              void* d_out, int out_size, void* d_ws, size_t ws_size,
                              hipStream_t stream)
{
  (void)in_sizes; (void)n_in; (void)out_size; (void)ws_size;

  const float* x     = (const float*)d_in[0];
  const float* Wqkv  = (const float*)d_in[1];
  const float* Wq    = (const float*)d_in[2];
  const float* Wkv   = (const float*)d_in[3];
  const float* Wproj = (const float*)d_in[4];
  const float* bproj = (const float*)d_in[5];

  float* ws   = (float*)d_ws;
  float* qkv  = ws;                                    // B*N*3C
  float* x4   = qkv  + (size_t)B_ * N_ * 3 * C_;       // B*S*F*C
  float* xd   = x4   + (size_t)B_ * S_ * F_ * C_;      // B*S*C
  float* q2   = xd   + (size_t)B_ * S_ * C_;           // B*S*C
  float* k2   = q2   + (size_t)B_ * S_ * C_;           // B*S*F*C
  float* outp = k2   + (size_t)B_ * S_ * F_ * C_;      // B*N*C (pre-proj)

  float* out_final = (float*)d_out;
  float* attn2     = out_final + (size_t)B_ * N_ * C_;

  auto launch_gemm = [&](const float* A, int lda, const float* Bm, int ldb,
                         float* Cm, int ldc, int M, int Nc, int K,
                         const float* bias) {
    const int tiles  = ((M + 31) / 32) * (Nc / 64);
    const int blocks = (tiles + 7) / 8;
    hipLaunchKernelGGL(gemm_wmma_f32, dim3(blocks), dim3(256), 0, stream,
                       A, lda, Bm, ldb, Cm, ldc, M, Nc, K, bias);
  };

  // 1. qkv = x @ Wqkv                        [3138 x 2304, K=768]
  launch_gemm(x, C_, Wqkv, 3 * C_, qkv, 3 * C_, B_ * N_, 3 * C_, C_, nullptr);

  // 2. CLS attention -> out_pre rows b*N.
  hipLaunchKernelGGL(cls_attn_kernel, dim3(B_ * H_), dim3(256), 0, stream,
                     qkv, outp);

  // 3. Fused spatial attention -> x4.
  hipLaunchKernelGGL(spatial_attn_kernel, dim3(S_ / 16, F_, B_ * H_), dim3(32),
                     0, stream, qkv, x4);

  // 4. xd = frame-diagonal of x4.
  {
    const size_t total = (size_t)B_ * S_ * C_;
    hipLaunchKernelGGL(gather_xd_kernel, dim3((unsigned)((total + 255) / 256)),
                       dim3(256), 0, stream, x4, xd);
  }

  // 5. q2 = xd @ Wq                          [3136 x 768, K=768]
  launch_gemm(xd, C_, Wq, C_, q2, C_, B_ * S_, C_, C_, nullptr);

  // 6. k2 = x4 @ Wkv[:, :C]  (v-half of kv2 is dead in the reference)
  launch_gemm(x4, C_, Wkv, 2 * C_, k2, C_, B_ * S_ * F_, C_, C_, nullptr);

  // 7. Frame attention: attn2 (direct to d_out slice) + out_pre rows 1+s.
  hipLaunchKernelGGL(second_attn_kernel, dim3(B_ * S_), dim3(96), 0, stream,
                     q2, k2, x4, outp, attn2);

  // 8. Final projection: d_out[0:B*N*C] = out_pre @ Wproj + bproj.
  launch_gemm(outp, C_, Wproj, C_, out_final, C_, B_ * N_, C_, C_, bproj);
}